// SSP_func_65730179498738
// MI455X (gfx1250) — compile-verified
//
#include <hip/hip_runtime.h>
#include <math.h>

// ---------------------------------------------------------------------------
// Problem geometry (from reference): bs=16, C=256, H*W=16384
// ---------------------------------------------------------------------------
#define BS      16
#define NCH     256
#define NPIX    16384
#define CTILES  16          // 256 channels / 16-channel WMMA tiles
#define NCHUNKS 8           // pixel chunks per (b, ctile)
#define CHUNK   2048        // NPIX / NCHUNKS
#define TOPK    12

// Workspace layout (float offsets)
#define MFG_OFF   ((size_t)0)                       // [BS][NPIX] fg mask (0/1 float)
#define MBG_OFF   (MFG_OFF + (size_t)BS*NPIX)       // [BS][NPIX] bg mask
#define PF_OFF    (MBG_OFF + (size_t)BS*NPIX)       // [BS][NPIX] pred_fg (destroyed by topk)
#define PB_OFF    (PF_OFF  + (size_t)BS*NPIX)       // [BS][NPIX] pred_bg (destroyed by topk)
#define CNT_OFF   (PB_OFF  + (size_t)BS*NPIX)       // [2][BS] counts as float
#define IDX_OFF   (CNT_OFF + 32)                    // int region: [2][BS][TOPK]
#define PARTF_OFF (IDX_OFF + 2*BS*TOPK)             // [BS][NCH][NCHUNKS] fg partial sums
#define PARTB_OFF (PARTF_OFF + (size_t)BS*NCH*NCHUNKS)

typedef __attribute__((ext_vector_type(2))) float v2f;
typedef __attribute__((ext_vector_type(8))) float v8f;

// ---------------------------------------------------------------------------
// Kernel 1: softmax, thresholds -> masks (as 0/1 floats), probs, counts
// grid = BS blocks, 256 threads
// ---------------------------------------------------------------------------
__global__ void ssp_prep_kernel(const float* __restrict__ out2,
                                const float* __restrict__ tau,
                                float* __restrict__ wsf) {
    const int b   = blockIdx.x;
    const int tid = threadIdx.x;

    const float t      = tau[0];
    const float fg_th  = 1.0f / (1.0f + expf(-t));
    const float bg_th  = 1.0f - fg_th;

    const float* o0 = out2 + ((size_t)b * 2 + 0) * NPIX;
    const float* o1 = out2 + ((size_t)b * 2 + 1) * NPIX;
    float* mfg = wsf + MFG_OFF + (size_t)b * NPIX;
    float* mbg = wsf + MBG_OFF + (size_t)b * NPIX;
    float* pf  = wsf + PF_OFF  + (size_t)b * NPIX;
    float* pb  = wsf + PB_OFF  + (size_t)b * NPIX;

    int lcf = 0, lcb = 0;
    for (int n = tid; n < NPIX; n += 256) {
        float a = o0[n], c = o1[n];
        float m  = fmaxf(a, c);
        float e0 = expf(a - m);
        float e1 = expf(c - m);
        float inv = 1.0f / (e0 + e1);
        float p1 = e1 * inv;            // pred_fg
        float p0 = e0 * inv;            // pred_bg
        float m1 = (p1 > fg_th) ? 1.0f : 0.0f;
        float m0 = (p0 > bg_th) ? 1.0f : 0.0f;
        mfg[n] = m1;  mbg[n] = m0;
        pf[n]  = p1;  pb[n]  = p0;
        lcf += (int)m1;  lcb += (int)m0;
    }

    __shared__ int scf[256], scb[256];
    scf[tid] = lcf; scb[tid] = lcb;
    __syncthreads();
    for (int s = 128; s > 0; s >>= 1) {
        if (tid < s) { scf[tid] += scf[tid + s]; scb[tid] += scb[tid + s]; }
        __syncthreads();
    }
    if (tid == 0) {
        wsf[CNT_OFF + b]      = (float)scf[0];
        wsf[CNT_OFF + BS + b] = (float)scb[0];
    }
}

// ---------------------------------------------------------------------------
// Kernel 2: top-12 indices per (class, batch). grid = 2*BS blocks, 256 thr.
// Iterative argmax with lowest-index tie-break (matches lax.top_k ordering).
// Destroys pf/pb (regenerated each call by prep -> replay-safe).
// ---------------------------------------------------------------------------
__global__ void ssp_topk_kernel(float* __restrict__ wsf) {
    const int bc  = blockIdx.x;
    const int cls = bc & 1;          // 0 = fg, 1 = bg
    const int b   = bc >> 1;
    const int tid = threadIdx.x;

    float* arr = wsf + (cls ? PB_OFF : PF_OFF) + (size_t)b * NPIX;
    int*   idx = ((int*)(wsf + IDX_OFF)) + (cls * BS + b) * TOPK;

    __shared__ float bv[256];
    __shared__ int   bi[256];

    for (int k = 0; k < TOPK; ++k) {
        float best  = -INFINITY;
        int   besti = 1 << 30;
        for (int n = tid; n < NPIX; n += 256) {
            float v = arr[n];
            if (v > best || (v == best && n < besti)) { best = v; besti = n; }
        }
        bv[tid] = best; bi[tid] = besti;
        __syncthreads();
        for (int s = 128; s > 0; s >>= 1) {
            if (tid < s) {
                float v2 = bv[tid + s]; int i2 = bi[tid + s];
                if (v2 > bv[tid] || (v2 == bv[tid] && i2 < bi[tid])) {
                    bv[tid] = v2; bi[tid] = i2;
                }
            }
            __syncthreads();
        }
        if (tid == 0) {
            idx[k] = bi[0];
            arr[bi[0]] = -INFINITY;   // exclude for next round
        }
        __syncthreads();
    }
}

// ---------------------------------------------------------------------------
// Kernel 3: main bandwidth kernel. grid = (NCHUNKS, CTILES, BS), 256 threads.
// Each wave streams a 16-channel x 256-pixel strip of feature_q and does
// D = A(16x4 fp32 feat) x B(4x16: col0=fg mask, col1=bg mask) + C
// via v_wmma_f32_16x16x4_f32.  fg sums land in C columns 0 (lanes 0/16),
// bg sums in column 1 (lanes 1/17).  Deterministic LDS reduce across waves.
// ---------------------------------------------------------------------------
__global__ void ssp_wmma_reduce_kernel(const float* __restrict__ feat,
                                       const float* __restrict__ wsf,
                                       float* __restrict__ wsout) {
    const int chunk = blockIdx.x;
    const int ct    = blockIdx.y;
    const int b     = blockIdx.z;

    const int tid  = threadIdx.x;
    const int lane = tid & 31;
    const int wv   = tid >> 5;          // 8 waves
    const int half = lane >> 4;         // 0: K=0..1, 1: K=2..3
    const int mrow = lane & 15;         // A: channel-in-tile, B: column

    const int n0 = chunk * CHUNK + wv * (CHUNK / 8);   // 256 pixels per wave

    // A fragment stream: lane holds feat[c = ct*16+mrow][n0 + 2*half + {0,1}]
    const float* ap = feat
        + ((size_t)b * NCH + (size_t)ct * 16 + mrow) * NPIX
        + n0 + half * 2;

    // B fragment stream: col0 -> fg mask, col1 -> bg mask, cols>=2 -> 0
    const float* mbase = (mrow == 0) ? (wsf + MFG_OFF) : (wsf + MBG_OFF);
    const float* bp = mbase + (size_t)b * NPIX + n0 + half * 2;
    const float  sel = (mrow < 2) ? 1.0f : 0.0f;

    v8f acc = {};
#pragma unroll 4
    for (int it = 0; it < (CHUNK / 8) / 4; ++it) {      // 64 K=4 steps
        v2f a = *(const v2f*)ap;  ap += 4;
        v2f m = *(const v2f*)bp;  bp += 4;
        v2f bvv = m * sel;
        acc = __builtin_amdgcn_wmma_f32_16x16x4_f32(
            /*neg_a=*/false, a, /*neg_b=*/false, bvv,
            /*c_mod=*/(short)0, acc, /*reuse_a=*/false, /*reuse_b=*/false);
    }

    // C/D layout: VGPR r: lanes0-15 -> M=r, N=lane; lanes16-31 -> M=8+r, N=lane-16
    __shared__ float red[8][2][16];
    if (lane == 0)  { for (int r = 0; r < 8; ++r) red[wv][0][r]     = acc[r]; }
    if (lane == 16) { for (int r = 0; r < 8; ++r) red[wv][0][8 + r] = acc[r]; }
    if (lane == 1)  { for (int r = 0; r < 8; ++r) red[wv][1][r]     = acc[r]; }
    if (lane == 17) { for (int r = 0; r < 8; ++r) red[wv][1][8 + r] = acc[r]; }
    __syncthreads();

    if (tid < 32) {
        const int cls = tid >> 4;
        const int m   = tid & 15;
        float s = 0.0f;
        for (int w = 0; w < 8; ++w) s += red[w][cls][m];   // fixed order
        const int c = ct * 16 + m;
        float* dst = wsout + (cls ? PARTB_OFF : PARTF_OFF);
        dst[((size_t)b * NCH + c) * NCHUNKS + chunk] = s;
    }
}

// ---------------------------------------------------------------------------
// Kernel 4: finalize. 16 blocks x 256 threads, one thread per (b, c).
// ---------------------------------------------------------------------------
__global__ void ssp_finalize_kernel(const float* __restrict__ feat,
                                    const float* __restrict__ wsf,
                                    float* __restrict__ outp) {
    const int gid = blockIdx.x * 256 + threadIdx.x;   // 0..4095
    const int b   = gid >> 8;
    const int c   = gid & 255;

    const float* pfp = wsf + PARTF_OFF + ((size_t)b * NCH + c) * NCHUNKS;
    const float* pbp = wsf + PARTB_OFF + ((size_t)b * NCH + c) * NCHUNKS;
    float sf = 0.0f, sb = 0.0f;
    for (int k = 0; k < NCHUNKS; ++k) { sf += pfp[k]; sb += pbp[k]; }

    const float cf = wsf[CNT_OFF + b];
    const float cb = wsf[CNT_OFF + BS + b];
    const int*  ib = (const int*)(wsf + IDX_OFF);
    const float* frow = feat + ((size_t)b * NCH + c) * NPIX;

    float rf, rb;
    if (cf > 0.0f) {
        rf = sf / fmaxf(cf, 1.0f);
    } else {
        const int* idx = ib + (0 * BS + b) * TOPK;
        float s = 0.0f;
        for (int k = 0; k < TOPK; ++k) s += frow[idx[k]];
        rf = s * (1.0f / TOPK);
    }
    if (cb > 0.0f) {
        rb = sb / fmaxf(cb, 1.0f);
    } else {
        const int* idx = ib + (1 * BS + b) * TOPK;
        float s = 0.0f;
        for (int k = 0; k < TOPK; ++k) s += frow[idx[k]];
        rb = s * (1.0f / TOPK);
    }

    outp[(size_t)b * NCH + c]                    = rf;   // fg_proto
    outp[(size_t)BS * NCH + (size_t)b * NCH + c] = rb;   // bg_proto
}

// ---------------------------------------------------------------------------
extern "C" void kernel_launch(void* const* d_in, const int* in_sizes, int n_in,
                              void* d_out, int out_size, void* d_ws, size_t ws_size,
                              hipStream_t stream) {
    const float* feat = (const float*)d_in[0];   // (16,256,128,128) fp32
    const float* out2 = (const float*)d_in[1];   // (16,2,128,128) fp32
    const float* tau  = (const float*)d_in[2];   // scalar fp32
    float* o   = (float*)d_out;                  // 2*16*256 floats (fg then bg)
    float* wsf = (float*)d_ws;                   // ~4.3 MB used

    ssp_prep_kernel<<<dim3(BS), dim3(256), 0, stream>>>(out2, tau, wsf);
    ssp_topk_kernel<<<dim3(2 * BS), dim3(256), 0, stream>>>(wsf);
    ssp_wmma_reduce_kernel<<<dim3(NCHUNKS, CTILES, BS), dim3(256), 0, stream>>>(feat, wsf, wsf);
    ssp_finalize_kernel<<<dim3(BS), dim3(256), 0, stream>>>(feat, wsf, o);
}